// MambaLayer_57303453663841
// MI455X (gfx1250) — compile-verified
//
#include <hip/hip_runtime.h>
#include <hip/hip_bf16.h>
#include <math.h>

// ---------------- problem dims (compile-time) ----------------
#define D_MODEL 768
#define D_STATE 16
#define D_CONV  4
#define D_INNER 1536          // EXPAND * D_MODEL
#define DT_RANK 48            // ceil(768/16)
#define B_SZ    2
#define L_SEQ   2048
#define NROW    (B_SZ * L_SEQ)            // 4096 token rows
#define XPROJ_N (DT_RANK + 2 * D_STATE)   // 80
#define KC      32                        // K-chunk staged per TDM transfer
#define NCH     (D_MODEL / KC)            // 24 chunks for gemm_in

typedef __attribute__((ext_vector_type(2))) float v2f;
typedef __attribute__((ext_vector_type(8))) float v8f;
typedef __attribute__((ext_vector_type(4))) unsigned int v4u;
typedef __attribute__((ext_vector_type(8))) int v8i;
typedef __attribute__((ext_vector_type(4))) int v4i;

// CDNA5 fp32 tensor op: D(16x16,f32) = A(16x4,f32) * B(4x16,f32) + C
__device__ __forceinline__ v8f wmma_f32_k4(v2f a, v2f b, v8f c) {
  return __builtin_amdgcn_wmma_f32_16x16x4_f32(false, a, false, b, (short)0, c,
                                               false, false);
}

// Register-blocked K-loop: MT 16-row tiles x NT 16-col tiles per wave.
// C[m,n] = sum_k A[m,k] * W[n,k]   (A row stride lda, W row stride ldw)
template <int MT, int NT>
__device__ __forceinline__ void gemm_acc_mt(const float* __restrict__ A, int lda,
                                            const float* __restrict__ W, int ldw,
                                            int K, int m0, int n0, int lane,
                                            v8f acc[MT][NT]) {
  const int half = lane >> 4, lm = lane & 15;
  for (int kk = 0; kk < K; kk += 4) {
    v2f a[MT];
#pragma unroll
    for (int i = 0; i < MT; ++i)
      a[i] = *(const v2f*)(A + (size_t)(m0 + 16 * i + lm) * lda + kk + 2 * half);
#pragma unroll
    for (int j = 0; j < NT; ++j) {
      v2f b = *(const v2f*)(W + (size_t)(n0 + 16 * j + lm) * ldw + kk + 2 * half);
#pragma unroll
      for (int i = 0; i < MT; ++i)
        acc[i][j] = wmma_f32_k4(a[i], b, acc[i][j]);
    }
  }
}

// ---------------- kernel 1: xz = x * W_in^T  (4096 x 3072, K=768) ----------------
// Weight tile (128 rows x KC) staged into LDS by the Tensor Data Mover,
// double-buffered; wave0 issues the D# and synchronizes via TENSORcnt.
__global__ void __launch_bounds__(256) k_gemm_in(const float* __restrict__ X,
                                                 const float* __restrict__ Win,
                                                 float* __restrict__ XZ) {
  __shared__ float lds_w[2][128 * KC];

  const int lane = threadIdx.x & 31, wave = threadIdx.x >> 5;
  const int half = lane >> 4, lm = lane & 15;
  const int mwave = wave >> 1, nwave = wave & 1;
  const int m0 = blockIdx.x * 128 + mwave * 32;
  const int nblk = blockIdx.y * 128;

  v8f acc[2][4] = {};

  // raw LDS byte offset of the staging buffer (AS(3) pointer value)
  typedef __attribute__((address_space(3))) void lds_void_t;
  const unsigned lds_base =
      (unsigned)(unsigned long long)(lds_void_t*)&lds_w[0][0];

  const float* __restrict__ arow0 = X + (size_t)(m0 + lm) * D_MODEL;
  const float* __restrict__ arow1 = X + (size_t)(m0 + 16 + lm) * D_MODEL;

  auto tdm_issue = [&](int chunk, int buf) {
    const unsigned long long ga = (unsigned long long)(const void*)(
        Win + (size_t)nblk * D_MODEL + chunk * KC);
    v4u g0;
    g0[0] = 1u;                                            // count=1 (valid D#)
    g0[1] = lds_base + (unsigned)buf * (128u * KC * 4u);   // lds_addr
    g0[2] = (unsigned)(ga & 0xffffffffull);                // global_addr lo
    g0[3] = (unsigned)((ga >> 32) & 0x1ffffffull) | (2u << 30);  // hi | type=2
    v8i g1;
    g1[0] = 0x20000;                          // data_size=2 (4 bytes)
    g1[1] = (int)((D_MODEL & 0xffff) << 16);  // tensor_dim0[15:0]
    g1[2] = (int)(((D_MODEL >> 16) & 0xffff) | (128u << 16));  // td0 hi|td1 lo
    g1[3] = (int)(KC << 16);                  // td1 hi=0 | tile_dim0=KC
    g1[4] = 128;                              // tile_dim1=128 rows, tile_dim2=0
    g1[5] = D_MODEL;                          // tensor_dim0_stride lo32
    g1[6] = 0;
    g1[7] = 0;
    v4i z4 = {0, 0, 0, 0};
    v8i z8 = {0, 0, 0, 0, 0, 0, 0, 0};
    __builtin_amdgcn_tensor_load_to_lds(g0, g1, z4, z4, z8, 0);
  };

  // prologue: stage chunk 0
  if (threadIdx.x < 32) {
    tdm_issue(0, 0);
    __builtin_amdgcn_s_wait_tensorcnt(0);
  }
  __syncthreads();

  for (int c = 0; c < NCH; ++c) {
    const int buf = c & 1;
    if (threadIdx.x < 32 && (c + 1) < NCH) tdm_issue(c + 1, buf ^ 1);

#pragma unroll 4
    for (int kkl = 0; kkl < KC; kkl += 4) {
      const int kg = c * KC + kkl;
      v2f a0 = *(const v2f*)(arow0 + kg + 2 * half);
      v2f a1 = *(const v2f*)(arow1 + kg + 2 * half);
#pragma unroll
      for (int j = 0; j < 4; ++j) {
        const int r = nwave * 64 + 16 * j + lm;
        v2f b = *(const v2f*)(&lds_w[buf][r * KC + kkl + 2 * half]);
        acc[0][j] = wmma_f32_k4(a0, b, acc[0][j]);
        acc[1][j] = wmma_f32_k4(a1, b, acc[1][j]);
      }
    }

    if (threadIdx.x < 32 && (c + 1) < NCH)
      __builtin_amdgcn_s_wait_tensorcnt(0);
    __syncthreads();
  }

#pragma unroll
  for (int i = 0; i < 2; ++i)
#pragma unroll
    for (int j = 0; j < 4; ++j)
#pragma unroll
      for (int ii = 0; ii < 8; ++ii)
        XZ[(size_t)(m0 + 16 * i + 8 * half + ii) * (2 * D_INNER) + nblk +
           nwave * 64 + 16 * j + lm] = acc[i][j][ii];
}

// ---------------- kernel 2: uc = silu(causal depthwise conv(u) + b) ----------------
__global__ void k_conv_silu(const float* __restrict__ XZ,
                            const float* __restrict__ cw,
                            const float* __restrict__ cb,
                            float* __restrict__ UC) {
  const int idx = blockIdx.x * blockDim.x + threadIdx.x;
  if (idx >= NROW * D_INNER) return;
  const int d   = idx % D_INNER;
  const int row = idx / D_INNER;   // b*L + l
  const int l   = row % L_SEQ;
  float s = cb[d];
#pragma unroll
  for (int k = 0; k < D_CONV; ++k) {
    const int lk = l - (D_CONV - 1) + k;
    if (lk >= 0)
      s += XZ[(size_t)(row + (lk - l)) * (2 * D_INNER) + d] *
           cw[d * D_CONV + k];
  }
  s = s / (1.0f + __expf(-s));   // silu
  UC[(size_t)row * D_INNER + d] = s;
}

// ---------------- kernel 3: xproj = uc * W_xproj^T  (4096 x 80, K=1536) ----------------
__global__ void __launch_bounds__(256) k_gemm_xproj(const float* __restrict__ UC,
                                                    const float* __restrict__ Wx,
                                                    float* __restrict__ XP) {
  const int lane = threadIdx.x & 31, wave = threadIdx.x >> 5;
  const int m0 = blockIdx.x * 128 + wave * 16;
  v8f acc[1][5] = {};
  gemm_acc_mt<1, 5>(UC, D_INNER, Wx, D_INNER, D_INNER, m0, 0, lane, acc);
  const int half = lane >> 4, lm = lane & 15;
#pragma unroll
  for (int j = 0; j < 5; ++j)
#pragma unroll
    for (int i = 0; i < 8; ++i)
      XP[(size_t)(m0 + 8 * half + i) * XPROJ_N + 16 * j + lm] = acc[0][j][i];
}

// ---------------- kernel 4: dt = softplus(dt_low * W_dt^T + b_dt)  (4096 x 1536, K=48) ----------------
__global__ void __launch_bounds__(256) k_gemm_dt(const float* __restrict__ XP,
                                                 const float* __restrict__ Wdt,
                                                 const float* __restrict__ bdt,
                                                 float* __restrict__ DT) {
  const int lane = threadIdx.x & 31, wave = threadIdx.x >> 5;
  const int mwave = wave >> 1, nwave = wave & 1;
  const int m0 = blockIdx.x * 128 + mwave * 32;
  const int n0 = blockIdx.y * 128 + nwave * 64;
  v8f acc[2][4] = {};
  gemm_acc_mt<2, 4>(XP, XPROJ_N, Wdt, DT_RANK, DT_RANK, m0, n0, lane, acc);
  const int half = lane >> 4, lm = lane & 15;
#pragma unroll
  for (int i = 0; i < 2; ++i)
#pragma unroll
    for (int j = 0; j < 4; ++j) {
      const int col = n0 + 16 * j + lm;
      const float b = bdt[col];
#pragma unroll
      for (int ii = 0; ii < 8; ++ii) {
        float v = acc[i][j][ii] + b;
        v = (v > 20.0f) ? v : log1pf(__expf(v));   // softplus
        DT[(size_t)(m0 + 16 * i + 8 * half + ii) * D_INNER + col] = v;
      }
    }
}

// ---------------- kernel 5: selective scan + fused D-skip and SiLU gate ----------------
// block = 256 threads = 16 channels x 16 states; grid = (D_INNER/16, B)
__global__ void k_scan(const float* __restrict__ DT,
                       const float* __restrict__ XP,
                       const float* __restrict__ UC,
                       const float* __restrict__ XZ,   // z = XZ[:, D_INNER:]
                       const float* __restrict__ Alog,
                       const float* __restrict__ Dvec,
                       float* __restrict__ Y) {
  const int t = threadIdx.x;
  const int d = blockIdx.x * 16 + (t >> 4);
  const int n = t & 15;
  const int b = blockIdx.y;

  const float A_dn = -__expf(Alog[d * D_STATE + n]);
  const float D_d  = Dvec[d];

  float h = 0.0f;
  for (int l = 0; l < L_SEQ; ++l) {
    const size_t row = (size_t)b * L_SEQ + l;

    // latency-bound recurrence: prefetch 8 steps ahead (global_prefetch_b8)
    if ((l & 7) == 0 && l + 8 < L_SEQ) {
      __builtin_prefetch(&DT[(row + 8) * D_INNER + d], 0, 1);
      __builtin_prefetch(&UC[(row + 8) * D_INNER + d], 0, 1);
      __builtin_prefetch(&XP[(row + 8) * XPROJ_N + DT_RANK + n], 0, 1);
    }

    const float dtv = DT[row * D_INNER + d];
    const float ucv = UC[row * D_INNER + d];
    const float Bv  = XP[row * XPROJ_N + DT_RANK + n];
    const float Cv  = XP[row * XPROJ_N + DT_RANK + D_STATE + n];

    const float dA = __expf(dtv * A_dn);
    h = dA * h + (dtv * ucv) * Bv;

    float p = h * Cv;
    // reduce over 16 states (stays inside each 16-lane half of the wave32)
    p += __shfl_xor(p, 1, 32);
    p += __shfl_xor(p, 2, 32);
    p += __shfl_xor(p, 4, 32);
    p += __shfl_xor(p, 8, 32);

    if (n == 0) {
      float y = p + ucv * D_d;
      const float zv = XZ[row * (2 * D_INNER) + D_INNER + d];
      y *= zv / (1.0f + __expf(-zv));   // y * silu(z)
      Y[row * D_INNER + d] = y;
    }
  }
}

// ---------------- kernel 6: out = y * W_out^T  (4096 x 768, K=1536) ----------------
__global__ void __launch_bounds__(256) k_gemm_out(const float* __restrict__ Y,
                                                  const float* __restrict__ Wout,
                                                  float* __restrict__ OUT) {
  const int lane = threadIdx.x & 31, wave = threadIdx.x >> 5;
  const int mwave = wave >> 1, nwave = wave & 1;
  const int m0 = blockIdx.x * 128 + mwave * 32;
  const int n0 = blockIdx.y * 128 + nwave * 64;
  v8f acc[2][4] = {};
  gemm_acc_mt<2, 4>(Y, D_INNER, Wout, D_INNER, D_INNER, m0, n0, lane, acc);
  const int half = lane >> 4, lm = lane & 15;
#pragma unroll
  for (int i = 0; i < 2; ++i)
#pragma unroll
    for (int j = 0; j < 4; ++j)
#pragma unroll
      for (int ii = 0; ii < 8; ++ii)
        OUT[(size_t)(m0 + 16 * i + 8 * half + ii) * D_MODEL + n0 + 16 * j +
            lm] = acc[i][j][ii];
}

// ---------------- host launch ----------------
extern "C" void kernel_launch(void* const* d_in, const int* in_sizes, int n_in,
                              void* d_out, int out_size, void* d_ws,
                              size_t ws_size, hipStream_t stream) {
  const float* x    = (const float*)d_in[0];
  const float* W_in = (const float*)d_in[1];
  const float* cw   = (const float*)d_in[2];
  const float* cb   = (const float*)d_in[3];
  const float* Wx   = (const float*)d_in[4];
  const float* Wdt  = (const float*)d_in[5];
  const float* bdt  = (const float*)d_in[6];
  const float* Alog = (const float*)d_in[7];
  const float* Dvec = (const float*)d_in[8];
  const float* Wout = (const float*)d_in[9];
  float* out = (float*)d_out;

  // workspace layout (floats)
  float* ws = (float*)d_ws;
  float* XZ = ws;                                   // 4096*3072
  float* UC = XZ + (size_t)NROW * 2 * D_INNER;      // 4096*1536
  float* XP = UC + (size_t)NROW * D_INNER;          // 4096*80
  float* DT = XP + (size_t)NROW * XPROJ_N;          // 4096*1536
  float* Y  = DT + (size_t)NROW * D_INNER;          // 4096*1536
  (void)ws_size; (void)in_sizes; (void)n_in; (void)out_size;

  k_gemm_in<<<dim3(NROW / 128, (2 * D_INNER) / 128), 256, 0, stream>>>(x, W_in, XZ);
  k_conv_silu<<<(NROW * D_INNER + 255) / 256, 256, 0, stream>>>(XZ, cw, cb, UC);
  k_gemm_xproj<<<dim3(NROW / 128), 256, 0, stream>>>(UC, Wx, XP);
  k_gemm_dt<<<dim3(NROW / 128, D_INNER / 128), 256, 0, stream>>>(XP, Wdt, bdt, DT);
  k_scan<<<dim3(D_INNER / 16, B_SZ), 256, 0, stream>>>(DT, XP, UC, XZ, Alog, Dvec, Y);
  k_gemm_out<<<dim3(NROW / 128, D_MODEL / 128), 256, 0, stream>>>(Y, Wout, out);
}